// TnoBlock2d_5231270167022
// MI455X (gfx1250) — compile-verified
//
#include <hip/hip_runtime.h>
#include <hip/hip_bf16.h>
#include <math.h>

#define EPSV 1e-5f

typedef float v2f __attribute__((ext_vector_type(2)));
typedef float v8f __attribute__((ext_vector_type(8)));
typedef int   v4i __attribute__((ext_vector_type(4)));

#if defined(__has_builtin)
#if __has_builtin(__builtin_amdgcn_global_load_async_to_lds_b128) && \
    __has_builtin(__builtin_amdgcn_s_wait_asynccnt)
#define TNO_ASYNC_LDS 1
#endif
#endif

#if defined(TNO_ASYNC_LDS)
#define TNO_ASYNC_COPY16(gp, lp)                                          \
  __builtin_amdgcn_global_load_async_to_lds_b128(                         \
      (__attribute__((address_space(1))) v4i*)(gp),                       \
      (__attribute__((address_space(3))) v4i*)(lp), 0, 0)
#endif

// Problem sizes (fixed by the reference)
#define NIMG 256   // B*C = 4*64
#define HWSZ 65536 // 256*256

// ---------------- ws layout (float offsets) ----------------
static constexpr long OFF_BYF_KQV = 0;        // [32][256]
static constexpr long OFF_BYF_MIX = 8192;     // [128][256]
static constexpr long OFF_CX_K    = 40960;    // [8][256]
static constexpr long OFF_SX_K    = 43008;
static constexpr long OFF_CX_M    = 45056;    // [32][256]
static constexpr long OFF_SX_M    = 53248;
static constexpr long OFF_CY_KQ   = 61440;    // [64][16]
static constexpr long OFF_SY_KQ   = 62464;
static constexpr long OFF_CY_V    = 63488;    // [256][16]
static constexpr long OFF_SY_V    = 67584;
static constexpr long OFF_CY_M    = 71680;    // [256][64]
static constexpr long OFF_SY_M    = 88064;
static constexpr long OFF_BX_KQ   = 104448;   // [16][64]
static constexpr long OFF_BX_V    = 105472;   // [16][256]
static constexpr long OFF_BX_M    = 109568;   // [64][256]
static constexpr long TOTAL_BASIS = 125952;
static constexpr long OFF_STATS   = 125952;   // 8 arrays x 256
static constexpr long OFF_W0 = 131072;                 // 64 MiB buffers
static constexpr long OFF_W1 = OFF_W0 + 16777216;
static constexpr long OFF_W2 = OFF_W1 + 16777216;
static constexpr long OFF_T  = OFF_W2 + 16777216;      // [256][128][256]
static constexpr long OFF_GM = OFF_T  + 8388608;       // [256][256][64]
static constexpr long OFF_GK = OFF_GM + 4194304;       // [256][64][16]
static constexpr long OFF_GQ = OFF_GK + 262144;
static constexpr long OFF_GV = OFF_GQ + 262144;        // [256][256][16]
static constexpr long OFF_FK = OFF_GV + 1048576;       // [256][16][8][2]
static constexpr long OFF_FQ = OFF_FK + 65536;
static constexpr long OFF_FV = OFF_FQ + 65536;
static constexpr long OFF_FM = OFF_FV + 65536;         // [256][64][32][2]
static constexpr long OFF_KO = OFF_FM + 1048576;       // [256][4096]
static constexpr long OFF_QO = OFF_KO + 1048576;
static constexpr long OFF_S  = OFF_QO + 1048576;       // [4][64][64]

// ---------------- basis init ----------------
__global__ void tno_init_basis(float* __restrict__ w) {
  const float PI2 = 6.2831853071795864769f;
  for (long g = (long)blockIdx.x * blockDim.x + threadIdx.x; g < TOTAL_BASIS;
       g += (long)gridDim.x * blockDim.x) {
    long i = g;
    if (i < 8192) { // BYF_KQV [32][256]: rows 0-15 cos, 16-31 -sin, f_r in {0..7,-8..-1}
      int r2 = (int)(i >> 8), y = (int)(i & 255);
      int r = r2 & 15;
      int f = (r < 8) ? r : r - 16;
      int m = ((f * y) % 256 + 256) % 256;
      float th = PI2 * (float)m * (1.0f / 256.0f);
      w[OFF_BYF_KQV + i] = (r2 < 16) ? cosf(th) : -sinf(th);
      continue;
    }
    i -= 8192;
    if (i < 32768) { // BYF_MIX [128][256]
      int r2 = (int)(i >> 8), y = (int)(i & 255);
      int r = r2 & 63;
      int f = (r < 32) ? r : r - 64;
      int m = ((f * y) % 256 + 256) % 256;
      float th = PI2 * (float)m * (1.0f / 256.0f);
      w[OFF_BYF_MIX + i] = (r2 < 64) ? cosf(th) : -sinf(th);
      continue;
    }
    i -= 32768;
    if (i < 4096) { // CX_K / SX_K [8][256]
      long j = i & 2047;
      int c = (int)(j >> 8), x = (int)(j & 255);
      int m = (c * x) & 255;
      float th = PI2 * (float)m * (1.0f / 256.0f);
      if (i < 2048) w[OFF_CX_K + j] = cosf(th); else w[OFF_SX_K + j] = sinf(th);
      continue;
    }
    i -= 4096;
    if (i < 16384) { // CX_M / SX_M [32][256]
      long j = i & 8191;
      int c = (int)(j >> 8), x = (int)(j & 255);
      int m = (c * x) & 255;
      float th = PI2 * (float)m * (1.0f / 256.0f);
      if (i < 8192) w[OFF_CX_M + j] = cosf(th); else w[OFF_SX_M + j] = sinf(th);
      continue;
    }
    i -= 16384;
    if (i < 2048) { // CY_KQ / SY_KQ [64][16], Ho=64
      long j = i & 1023;
      int p = (int)(j >> 4), r = (int)(j & 15);
      int f = (r < 8) ? r : r - 16;
      int m = ((f * p) % 64 + 64) % 64;
      float th = PI2 * (float)m * (1.0f / 64.0f);
      if (i < 1024) w[OFF_CY_KQ + j] = cosf(th); else w[OFF_SY_KQ + j] = sinf(th);
      continue;
    }
    i -= 2048;
    if (i < 8192) { // CY_V / SY_V [256][16], Ho=256
      long j = i & 4095;
      int p = (int)(j >> 4), r = (int)(j & 15);
      int f = (r < 8) ? r : r - 16;
      int m = ((f * p) % 256 + 256) % 256;
      float th = PI2 * (float)m * (1.0f / 256.0f);
      if (i < 4096) w[OFF_CY_V + j] = cosf(th); else w[OFF_SY_V + j] = sinf(th);
      continue;
    }
    i -= 8192;
    if (i < 32768) { // CY_M / SY_M [256][64]
      long j = i & 16383;
      int p = (int)(j >> 6), r = (int)(j & 63);
      int f = (r < 32) ? r : r - 64;
      int m = ((f * p) % 256 + 256) % 256;
      float th = PI2 * (float)m * (1.0f / 256.0f);
      if (i < 16384) w[OFF_CY_M + j] = cosf(th); else w[OFF_SY_M + j] = sinf(th);
      continue;
    }
    i -= 32768;
    if (i < 1024) { // BX_KQ [16][64], Wo=64
      int j = (int)(i >> 6), q = (int)(i & 63);
      int c = j >> 1;
      float cw = (c == 0) ? 1.0f : 2.0f;
      int m = (c * q) % 64;
      float th = PI2 * (float)m * (1.0f / 64.0f);
      w[OFF_BX_KQ + i] = (j & 1) ? -cw * sinf(th) : cw * cosf(th);
      continue;
    }
    i -= 1024;
    if (i < 4096) { // BX_V [16][256], Wo=256
      int j = (int)(i >> 8), q = (int)(i & 255);
      int c = j >> 1;
      float cw = (c == 0) ? 1.0f : 2.0f;
      int m = (c * q) & 255;
      float th = PI2 * (float)m * (1.0f / 256.0f);
      w[OFF_BX_V + i] = (j & 1) ? -cw * sinf(th) : cw * cosf(th);
      continue;
    }
    i -= 4096;
    { // BX_M [64][256]
      int j = (int)(i >> 8), q = (int)(i & 255);
      int c = j >> 1;
      float cw = (c == 0) ? 1.0f : 2.0f;
      int m = (c * q) & 255;
      float th = PI2 * (float)m * (1.0f / 256.0f);
      w[OFF_BX_M + i] = (j & 1) ? -cw * sinf(th) : cw * cosf(th);
    }
  }
}

// ---------------- per-image mean/var over 65536 elements ----------------
__global__ void tno_stats(const float* __restrict__ x, float* __restrict__ mean,
                          float* __restrict__ var) {
  int n = blockIdx.x;
  const float* p = x + (long)n * HWSZ;
  float s = 0.f, s2 = 0.f;
  for (int i = threadIdx.x; i < HWSZ; i += 256) { float v = p[i]; s += v; s2 += v * v; }
  __shared__ float sh[256], sh2[256];
  sh[threadIdx.x] = s; sh2[threadIdx.x] = s2; __syncthreads();
  for (int o = 128; o > 0; o >>= 1) {
    if (threadIdx.x < o) { sh[threadIdx.x] += sh[threadIdx.x + o]; sh2[threadIdx.x] += sh2[threadIdx.x + o]; }
    __syncthreads();
  }
  if (threadIdx.x == 0) {
    float m = sh[0] * (1.0f / HWSZ);
    mean[n] = m;
    var[n] = fmaxf(sh2[0] * (1.0f / HWSZ) - m * m, 0.0f);
  }
}

// ---------------- elementwise kernels ----------------
__global__ void tno_norm_ew(float* __restrict__ dst, const float* __restrict__ src,
                            const float* __restrict__ mean, const float* __restrict__ var,
                            const float* __restrict__ w, const float* __restrict__ b) {
  long i = (long)blockIdx.x * 256 + threadIdx.x;
  int n = (int)(i >> 16);
  float rs = rsqrtf(var[n] + EPSV);
  dst[i] = (src[i] - mean[n]) * rs * w[0] + b[0];
}

__global__ void tno_norm_add_ew(float* __restrict__ dst, const float* __restrict__ src,
                                const float* __restrict__ res,
                                const float* __restrict__ mean, const float* __restrict__ var,
                                const float* __restrict__ w, const float* __restrict__ b) {
  long i = (long)blockIdx.x * 256 + threadIdx.x;
  int n = (int)(i >> 16);
  float rs = rsqrtf(var[n] + EPSV);
  dst[i] = (src[i] - mean[n]) * rs * w[0] + b[0] + res[i];
}

__global__ void tno_mixskip_ew(float* __restrict__ z, const float* __restrict__ attn,
                               const float* __restrict__ mbias, const float* __restrict__ sw,
                               const float* __restrict__ sb) {
  long i = (long)blockIdx.x * 256 + threadIdx.x;
  z[i] = z[i] + mbias[0] + attn[i] * sw[0] + sb[0];
}

// out = inorm(inorm(z; mnw,mnb); mow,mob) + att   (inner norm stats collapse analytically)
__global__ void tno_final_ew(float* __restrict__ out, const float* __restrict__ z,
                             const float* __restrict__ att,
                             const float* __restrict__ mean, const float* __restrict__ var,
                             const float* __restrict__ mnw, const float* __restrict__ mnb,
                             const float* __restrict__ mow, const float* __restrict__ mob) {
  long i = (long)blockIdx.x * 256 + threadIdx.x;
  int n = (int)(i >> 16);
  float v = var[n];
  float rs1 = rsqrtf(v + EPSV);
  float w1 = mnw[0];
  float v1 = w1 * w1 * v * rs1 * rs1;          // var of inner-normed signal
  float rs2 = rsqrtf(v1 + EPSV);
  float scale = rs1 * w1 * rs2 * mow[0];
  out[i] = (z[i] - mean[n]) * scale + mob[0] + att[i];
  (void)mnb;
}

// ---------------- single-wave WMMA GEMM (used for M=32 case) ----------------
__global__ void tno_gemm_wmma(const float* __restrict__ A, const float* __restrict__ B,
                              float* __restrict__ C, int K,
                              int lda, int ldb, int ldc,
                              long sA, long sB, long sC,
                              int transB, float alpha) {
  const int m0 = blockIdx.x << 4;
  const int n0 = blockIdx.y << 6;
  A += (long)blockIdx.z * sA;
  B += (long)blockIdx.z * sB;
  C += (long)blockIdx.z * sC;
  const int l  = threadIdx.x;
  const int mr = l & 15;
  const int kq = (l >> 4) << 1;  // 0 or 2
  v8f acc0 = {0.f,0.f,0.f,0.f,0.f,0.f,0.f,0.f};
  v8f acc1 = acc0, acc2 = acc0, acc3 = acc0;
  for (int k0 = 0; k0 < K; k0 += 4) {
    v2f a;
    {
      const float* ap = A + (long)(m0 + mr) * lda + (k0 + kq);
      a.x = ap[0]; a.y = ap[1];
    }
    v2f b0, b1, b2, b3;
    if (!transB) {
      const float* bp = B + (long)(k0 + kq) * ldb + n0 + mr;
      b0.x = bp[0];  b0.y = bp[ldb];
      b1.x = bp[16]; b1.y = bp[ldb + 16];
      b2.x = bp[32]; b2.y = bp[ldb + 32];
      b3.x = bp[48]; b3.y = bp[ldb + 48];
    } else {
      const float* bp = B + (long)(n0 + mr) * ldb + (k0 + kq);
      const long st = (long)ldb * 16;
      b0.x = bp[0];          b0.y = bp[1];
      b1.x = bp[st];         b1.y = bp[st + 1];
      b2.x = bp[2 * st];     b2.y = bp[2 * st + 1];
      b3.x = bp[3 * st];     b3.y = bp[3 * st + 1];
    }
    acc0 = __builtin_amdgcn_wmma_f32_16x16x4_f32(false, a, false, b0, (short)0, acc0, false, false);
    acc1 = __builtin_amdgcn_wmma_f32_16x16x4_f32(false, a, false, b1, (short)0, acc1, false, false);
    acc2 = __builtin_amdgcn_wmma_f32_16x16x4_f32(false, a, false, b2, (short)0, acc2, false, false);
    acc3 = __builtin_amdgcn_wmma_f32_16x16x4_f32(false, a, false, b3, (short)0, acc3, false, false);
  }
  const int ro = (l >> 4) << 3;
  for (int i = 0; i < 8; ++i) {
    float* cp = C + (long)(m0 + ro + i) * ldc + n0 + mr;
    cp[0]  = acc0[i] * alpha;
    cp[16] = acc1[i] * alpha;
    cp[32] = acc2[i] * alpha;
    cp[48] = acc3[i] * alpha;
  }
}

// ---------------- workgroup WMMA GEMM: 4 waves, 64x64 C tile, B staged in LDS ----------
// B tile (16 k-rows x 64 cols, or 64 n-rows x 16 k for transB) is loaded once per block
// (async global->LDS when available), double-buffered, shared by the 4 M-strip waves.
__global__ void __launch_bounds__(128)
tno_gemm_wmma_wg(const float* __restrict__ A, const float* __restrict__ B,
                 float* __restrict__ C, int K,
                 int lda, int ldb, int ldc,
                 long sA, long sB, long sC,
                 int transB, float alpha) {
  __shared__ float bs[2][16 * 64];
  const int tid  = threadIdx.x;        // 0..127
  const int wave = tid >> 5;
  const int l    = tid & 31;
  const int mr   = l & 15;
  const int kq   = (l >> 4) << 1;      // 0 or 2
  const int m0   = (blockIdx.x << 6) + (wave << 4);
  const int n0   = blockIdx.y << 6;
  A += (long)blockIdx.z * sA;
  B += (long)blockIdx.z * sB;
  C += (long)blockIdx.z * sC;

  // staging geometry: each thread moves 8 consecutive floats (2x b128)
  int srow, scol, lofs;
  if (!transB) { srow = tid >> 3; scol = (tid & 7) * 8; lofs = srow * 64 + scol; }
  else         { srow = tid >> 1; scol = (tid & 1) * 8; lofs = srow * 16 + scol; }

  v8f acc0 = {0.f,0.f,0.f,0.f,0.f,0.f,0.f,0.f};
  v8f acc1 = acc0, acc2 = acc0, acc3 = acc0;

  const int nchunk = K >> 4;

  // stage chunk 0 into buffer 0
  {
    const float* gp = (!transB) ? (B + (long)srow * ldb + n0 + scol)
                                : (B + (long)(n0 + srow) * ldb + scol);
    float* lp = &bs[0][lofs];
#if defined(TNO_ASYNC_LDS)
    TNO_ASYNC_COPY16(gp, lp);
    TNO_ASYNC_COPY16(gp + 4, lp + 4);
#else
    float4 va = *(const float4*)gp;
    float4 vb = *(const float4*)(gp + 4);
    *(float4*)lp = va;
    *(float4*)(lp + 4) = vb;
#endif
  }

  for (int ch = 0; ch < nchunk; ++ch) {
#if defined(TNO_ASYNC_LDS)
    __builtin_amdgcn_s_wait_asynccnt(0);
#endif
    __syncthreads();
    // prefetch next chunk into the other buffer (overlaps with WMMA below)
    if (ch + 1 < nchunk) {
      int k1 = (ch + 1) << 4;
      const float* gp = (!transB) ? (B + (long)(k1 + srow) * ldb + n0 + scol)
                                  : (B + (long)(n0 + srow) * ldb + k1 + scol);
      float* lp = &bs[(ch + 1) & 1][lofs];
#if defined(TNO_ASYNC_LDS)
      TNO_ASYNC_COPY16(gp, lp);
      TNO_ASYNC_COPY16(gp + 4, lp + 4);
#else
      float4 va = *(const float4*)gp;
      float4 vb = *(const float4*)(gp + 4);
      *(float4*)lp = va;
      *(float4*)(lp + 4) = vb;
#endif
    }
    // compute 4 WMMA k-steps from the current LDS buffer
    const float* ls = bs[ch & 1];
    const int k0 = ch << 4;
    for (int kk = 0; kk < 16; kk += 4) {
      v2f a;
      const float* ap = A + (long)(m0 + mr) * lda + (k0 + kk + kq);
      a.x = ap[0]; a.y = ap[1];
      v2f b0, b1, b2, b3;
      if (!transB) {
        const float* bp = ls + (kk + kq) * 64 + mr;
        b0.x = bp[0];  b0.y = bp[64];
        b1.x = bp[16]; b1.y = bp[80];
        b2.x = bp[32]; b2.y = bp[96];
        b3.x = bp[48]; b3.y = bp[112];
      } else {
        const float* bp = ls + mr * 16 + kk + kq;
        b0.x = bp[0];   b0.y = bp[1];
        b1.x = bp[256]; b1.y = bp[257];
        b2.x = bp[512]; b2.y = bp[513];
        b3.x = bp[768]; b3.y = bp[769];
      }
      acc0 = __builtin_amdgcn_wmma_f32_16x16x4_f32(false, a, false, b0, (short)0, acc0, false, false);
      acc1 = __builtin_amdgcn_wmma_f32_16x16x4_f32(false, a, false, b1, (short)0, acc1, false, false);
      acc2 = __builtin_amdgcn_wmma_f32_16x16x4_f32(false, a, false, b2, (short)0, acc2, false, false);
      acc3 = __builtin_amdgcn_wmma_f32_16x16x4_f32(false, a, false, b3, (short)0, acc3, false, false);
    }
    __syncthreads();
  }

  const int ro = (l >> 4) << 3;
  for (int i = 0; i < 8; ++i) {
    float* cp = C + (long)(m0 + ro + i) * ldc + n0 + mr;
    cp[0]  = acc0[i] * alpha;
    cp[16] = acc1[i] * alpha;
    cp[32] = acc2[i] * alpha;
    cp[48] = acc3[i] * alpha;
  }
}

// ---------------- mode-space stage: x-transform + weight multiply (kqv) ----------------
__global__ void tno_modes_kqv(const float* __restrict__ T,
                              const float* __restrict__ CX, const float* __restrict__ SX,
                              const float* __restrict__ kw1, const float* __restrict__ kw2,
                              const float* __restrict__ qw1, const float* __restrict__ qw2,
                              const float* __restrict__ vw1, const float* __restrict__ vw2,
                              float* __restrict__ Fk, float* __restrict__ Fq,
                              float* __restrict__ Fv) {
  int n = blockIdx.x;
  int t = threadIdx.x;           // 0..127
  int r = t >> 3, c = t & 7;
  const float* tc = T + ((long)n * 32 + r) * 256;
  const float* ts = tc + 16 * 256;
  const float* cx = CX + c * 256;
  const float* sx = SX + c * 256;
  float xr = 0.f, xi = 0.f;
  for (int x = 0; x < 256; ++x) {
    float av = tc[x], bv = ts[x], cv = cx[x], sv = sx[x];
    xr += av * cv + bv * sv;
    xi += bv * cv - av * sv;
  }
  const float inv = 1.0f / 65536.0f;
  xr *= inv; xi *= inv;
  int wi = ((r < 8) ? r : (r - 8)) * 8 + c;
  long o = (((long)n * 16 + r) * 8 + c) * 2;
  {
    const float* wp = (r < 8) ? kw1 : kw2;
    float wr = wp[wi * 2], wim = wp[wi * 2 + 1];
    Fk[o] = xr * wr - xi * wim; Fk[o + 1] = xr * wim + xi * wr;
  }
  {
    const float* wp = (r < 8) ? qw1 : qw2;
    float wr = wp[wi * 2], wim = wp[wi * 2 + 1];
    Fq[o] = xr * wr - xi * wim; Fq[o + 1] = xr * wim + xi * wr;
  }
  {
    const float* wp = (r < 8) ? vw1 : vw2;
    float wr = wp[wi * 2], wim = wp[wi * 2 + 1];
    Fv[o] = xr * wr - xi * wim; Fv[o + 1] = xr * wim + xi * wr;
  }
}

// mix variant: T [N][128][256], output Fm [N][64][32][2]
__global__ void tno_modes_mix(const float* __restrict__ T,
                              const float* __restrict__ CX, const float* __restrict__ SX,
                              const float* __restrict__ w1, const float* __restrict__ w2,
                              float* __restrict__ Fm) {
  int n = blockIdx.x;
  int t = blockIdx.y * 256 + threadIdx.x;  // 0..2047
  int r = t >> 5, c = t & 31;
  const float* tc = T + ((long)n * 128 + r) * 256;
  const float* ts = tc + 64 * 256;
  const float* cx = CX + c * 256;
  const float* sx = SX + c * 256;
  float xr = 0.f, xi = 0.f;
  for (int x = 0; x < 256; ++x) {
    float av = tc[x], bv = ts[x], cv = cx[x], sv = sx[x];
    xr += av * cv + bv * sv;
    xi += bv * cv - av * sv;
  }
  const float inv = 1.0f / 65536.0f;
  xr *= inv; xi *= inv;
  const float* wp = (r < 32) ? (w1 + ((r)*32 + c) * 2) : (w2 + ((r - 32) * 32 + c) * 2);
  float wr = wp[0], wim = wp[1];
  long o = (((long)n * 64 + r) * 32 + c) * 2;
  Fm[o] = xr * wr - xi * wim;
  Fm[o + 1] = xr * wim + xi * wr;
}

// ---------------- inverse y-stage: G[p][2c]=A, G[p][2c+1]=B ----------------
__global__ void tno_gstage(const float* __restrict__ F, const float* __restrict__ CY,
                           const float* __restrict__ SY, float* __restrict__ G,
                           int R, int Mh, int Ho) {
  int n = blockIdx.x;
  int total = Ho * Mh;
  for (int t = blockIdx.y * 256 + threadIdx.x; t < total; t += gridDim.y * 256) {
    int p = t / Mh, c = t % Mh;
    const float* f = F + (long)n * R * Mh * 2 + c * 2;
    const float* cy = CY + (long)p * R;
    const float* sy = SY + (long)p * R;
    float Av = 0.f, Bv = 0.f;
    for (int r = 0; r < R; ++r) {
      float fr = f[(long)r * Mh * 2], fi = f[(long)r * Mh * 2 + 1];
      float cv = cy[r], sv = sy[r];
      Av += fr * cv - fi * sv;
      Bv += fr * sv + fi * cv;
    }
    long o = ((long)n * Ho + p) * (2 * Mh) + 2 * c;
    G[o] = Av;
    G[o + 1] = Bv;
  }
}

// ---------------- softmax over rows of 64 ----------------
__global__ void tno_softmax(float* __restrict__ S) {
  int row = blockIdx.x;
  float* p = S + (long)row * 64;
  int t = threadIdx.x;  // 64 threads
  __shared__ float sh[64];
  float v = p[t];
  sh[t] = v; __syncthreads();
  for (int o = 32; o > 0; o >>= 1) { if (t < o) sh[t] = fmaxf(sh[t], sh[t + o]); __syncthreads(); }
  float mx = sh[0]; __syncthreads();
  float e = expf(v - mx);
  sh[t] = e; __syncthreads();
  for (int o = 32; o > 0; o >>= 1) { if (t < o) sh[t] += sh[t + o]; __syncthreads(); }
  p[t] = e / sh[0];
}

// ---------------- host orchestration ----------------
extern "C" void kernel_launch(void* const* d_in, const int* in_sizes, int n_in,
                              void* d_out, int out_size, void* d_ws, size_t ws_size,
                              hipStream_t stream) {
  (void)in_sizes; (void)n_in; (void)out_size; (void)ws_size;
  const float* x    = (const float*)d_in[0];
  const float* kw1  = (const float*)d_in[1];
  const float* kw2  = (const float*)d_in[2];
  const float* qw1  = (const float*)d_in[3];
  const float* qw2  = (const float*)d_in[4];
  const float* vw1  = (const float*)d_in[5];
  const float* vw2  = (const float*)d_in[6];
  const float* mw1  = (const float*)d_in[7];
  const float* mw2  = (const float*)d_in[8];
  const float* mbias= (const float*)d_in[9];
  const float* skw  = (const float*)d_in[10];
  const float* skb  = (const float*)d_in[11];
  const float* mnw  = (const float*)d_in[12];
  const float* mnb  = (const float*)d_in[13];
  const float* n1w  = (const float*)d_in[14];
  const float* n1b  = (const float*)d_in[15];
  const float* anw  = (const float*)d_in[16];
  const float* anb  = (const float*)d_in[17];
  const float* n2w  = (const float*)d_in[18];
  const float* n2b  = (const float*)d_in[19];
  const float* mow  = (const float*)d_in[20];
  const float* mob  = (const float*)d_in[21];
  float* out = (float*)d_out;
  float* w = (float*)d_ws;

  float* BYF_K = w + OFF_BYF_KQV; float* BYF_M = w + OFF_BYF_MIX;
  float* CXK = w + OFF_CX_K; float* SXK = w + OFF_SX_K;
  float* CXM = w + OFF_CX_M; float* SXM = w + OFF_SX_M;
  float* CYKQ = w + OFF_CY_KQ; float* SYKQ = w + OFF_SY_KQ;
  float* CYV = w + OFF_CY_V; float* SYV = w + OFF_SY_V;
  float* CYM = w + OFF_CY_M; float* SYM = w + OFF_SY_M;
  float* BXKQ = w + OFF_BX_KQ; float* BXV = w + OFF_BX_V; float* BXM = w + OFF_BX_M;
  float* meanX = w + OFF_STATS;       float* varX = meanX + 256;
  float* meanA = varX + 256;          float* varA = meanA + 256;
  float* meanT = varA + 256;          float* varT = meanT + 256;
  float* meanM = varT + 256;          float* varM = meanM + 256;
  float* W0 = w + OFF_W0;  // xa_n -> att_n
  float* W1 = w + OFF_W1;  // v -> mix(+skip)
  float* W2 = w + OFF_W2;  // att_raw -> att
  float* T  = w + OFF_T;
  float* GM = w + OFF_GM; float* GK = w + OFF_GK; float* GQ = w + OFF_GQ; float* GV = w + OFF_GV;
  float* FK = w + OFF_FK; float* FQ = w + OFF_FQ; float* FV = w + OFF_FV; float* FM = w + OFF_FM;
  float* KO = w + OFF_KO; float* QO = w + OFF_QO; float* S = w + OFF_S;

  // 0) trig bases
  tno_init_basis<<<256, 256, 0, stream>>>(w);
  // 1) xa_n = inorm(x)
  tno_stats<<<NIMG, 256, 0, stream>>>(x, meanX, varX);
  tno_norm_ew<<<65536, 256, 0, stream>>>(W0, x, meanX, varX, n1w, n1b);
  // 2) forward y-transform (kqv): T[n][32][256] = BYF_K @ xa_n[n]   (M=32 -> single-wave kernel)
  tno_gemm_wmma<<<dim3(2, 4, NIMG), 32, 0, stream>>>(BYF_K, W0, T, 256, 256, 256, 256,
                                                     0L, 65536L, 8192L, 0, 1.0f);
  // 3) x-transform + weight multiply -> Fk/Fq/Fv
  tno_modes_kqv<<<NIMG, 128, 0, stream>>>(T, CXK, SXK, kw1, kw2, qw1, qw2, vw1, vw2, FK, FQ, FV);
  // 4) inverse y-stage
  tno_gstage<<<dim3(NIMG, 2), 256, 0, stream>>>(FK, CYKQ, SYKQ, GK, 16, 8, 64);
  tno_gstage<<<dim3(NIMG, 2), 256, 0, stream>>>(FQ, CYKQ, SYKQ, GQ, 16, 8, 64);
  tno_gstage<<<dim3(NIMG, 8), 256, 0, stream>>>(FV, CYV, SYV, GV, 16, 8, 256);
  // 5) inverse x-stage: k,q at 64x64 ; v at 256x256 (workgroup WMMA kernel)
  tno_gemm_wmma_wg<<<dim3(1, 1, NIMG), 128, 0, stream>>>(GK, BXKQ, KO, 16, 16, 64, 64,
                                                         1024L, 0L, 4096L, 0, 1.0f);
  tno_gemm_wmma_wg<<<dim3(1, 1, NIMG), 128, 0, stream>>>(GQ, BXKQ, QO, 16, 16, 64, 64,
                                                         1024L, 0L, 4096L, 0, 1.0f);
  tno_gemm_wmma_wg<<<dim3(4, 4, NIMG), 128, 0, stream>>>(GV, BXV, W1, 16, 16, 256, 256,
                                                         4096L, 0L, 65536L, 0, 1.0f);
  // 6) attention: S = softmax(q k^T * 64); att_raw = S v
  tno_gemm_wmma_wg<<<dim3(1, 1, 4), 128, 0, stream>>>(QO, KO, S, 4096, 4096, 4096, 64,
                                                      262144L, 262144L, 4096L, 1, 64.0f);
  tno_softmax<<<256, 64, 0, stream>>>(S);
  tno_gemm_wmma_wg<<<dim3(1, 1024, 4), 128, 0, stream>>>(S, W1, W2, 64, 64, 65536, 65536,
                                                         4096L, 4194304L, 4194304L, 0, 1.0f);
  // 7) att = inorm(att_raw) + x  (in place in W2)
  tno_stats<<<NIMG, 256, 0, stream>>>(W2, meanA, varA);
  tno_norm_add_ew<<<65536, 256, 0, stream>>>(W2, W2, x, meanA, varA, anw, anb);
  // 8) att_n = inorm(att) -> W0
  tno_stats<<<NIMG, 256, 0, stream>>>(W2, meanT, varT);
  tno_norm_ew<<<65536, 256, 0, stream>>>(W0, W2, meanT, varT, n2w, n2b);
  // 9) mixer spectral conv on att_n
  tno_gemm_wmma_wg<<<dim3(2, 4, NIMG), 128, 0, stream>>>(BYF_M, W0, T, 256, 256, 256, 256,
                                                         0L, 65536L, 32768L, 0, 1.0f);
  tno_modes_mix<<<dim3(NIMG, 8), 256, 0, stream>>>(T, CXM, SXM, mw1, mw2, FM);
  tno_gstage<<<dim3(NIMG, 32), 256, 0, stream>>>(FM, CYM, SYM, GM, 64, 32, 256);
  tno_gemm_wmma_wg<<<dim3(4, 4, NIMG), 128, 0, stream>>>(GM, BXM, W1, 64, 64, 256, 256,
                                                         16384L, 0L, 65536L, 0, 1.0f);
  // 10) z = mix + bias + skip(att_n)   (in place in W1)
  tno_mixskip_ew<<<65536, 256, 0, stream>>>(W1, W0, mbias, skw, skb);
  // 11) out = inorm(inorm(z)) + att
  tno_stats<<<NIMG, 256, 0, stream>>>(W1, meanM, varM);
  tno_final_ew<<<65536, 256, 0, stream>>>(out, W1, W2, meanM, varM, mnw, mnb, mow, mob);
}